// DeformationGraph_61942018343649
// MI455X (gfx1250) — compile-verified
//
#include <hip/hip_runtime.h>

#define N_PTS    65536
#define N_NODES  4096
#define K_NN     20
#define D_ROW    35            // 3 + D_PE
#define DIS_THRESH 0.00021f
#define FAR_VAL  1000000000.0f

typedef __attribute__((ext_vector_type(2))) float v2f;
typedef __attribute__((ext_vector_type(8))) float v8f;

// Pack node data for the WMMA B operand: {x, y, z, -0.5*||v||^2}.
// Folding the node norm into the K=3 slot (paired with a constant 1.0 on the
// A side) makes each WMMA produce dot' = p.v - 0.5*||v||^2, so
// d2 = fma(dot', -2, ||p||^2) costs a single VALU op per candidate.
__global__ void pack_nodes_kernel(const float* __restrict__ vd,
                                  float4* __restrict__ out) {
    int j = blockIdx.x * blockDim.x + threadIdx.x;
    if (j < N_NODES) {
        float x = vd[3*j+0], y = vd[3*j+1], z = vd[3*j+2];
        out[j] = make_float4(x, y, z, -0.5f * (x*x + y*y + z*z));
    }
}

__launch_bounds__(256)
__global__ void deform_topk_blend_kernel(const float* __restrict__ inputs,
                                         const float4* __restrict__ nodes,
                                         const float* __restrict__ Rm,
                                         const float* __restrict__ g,
                                         const float* __restrict__ t,
                                         float* __restrict__ out_p,
                                         float* __restrict__ out_R) {
    // Per-wave staging buffer for one 32-point x 16-node d2 tile.
    // Row stride 17 (odd) -> conflict-free column writes / row reads.
    __shared__ float buf[8][32 * 17];

    const int tid      = threadIdx.x;
    const int lane     = tid & 31;
    const int wave     = tid >> 5;
    const int waveBase = blockIdx.x * 256 + wave * 32;
    const int myPt     = waveBase + lane;

    // This lane's own point (used for d2 combine + blend phase).
    const float px = inputs[myPt * D_ROW + 0];
    const float py = inputs[myPt * D_ROW + 1];
    const float pz = inputs[myPt * D_ROW + 2];
    const float p2 = px*px + py*py + pz*pz;

    // Build the two loop-invariant A operands (16x4 f32, M x K):
    // lanes 0-15 hold K=0,1 (x,y); lanes 16-31 hold K=2,3 (z, 1.0).
    const bool hi = (lane >= 16);
    const int  nn = lane & 15;
    v2f aLo, aHi;
    {
        const int q0 = waveBase + nn;
        const int q1 = q0 + 16;
        float x0 = inputs[q0*D_ROW+0], y0 = inputs[q0*D_ROW+1], z0 = inputs[q0*D_ROW+2];
        float x1 = inputs[q1*D_ROW+0], y1 = inputs[q1*D_ROW+1], z1 = inputs[q1*D_ROW+2];
        aLo[0] = hi ? z0 : x0;   aLo[1] = hi ? 1.0f : y0;
        aHi[0] = hi ? z1 : x1;   aHi[1] = hi ? 1.0f : y1;
    }

    // Sorted (ascending) top-K in registers.
    float best[K_NN];
    int   bidx[K_NN];
#pragma unroll
    for (int k = 0; k < K_NN; ++k) { best[k] = 3.0e38f; bidx[k] = 0; }

    // Guarded bubble-insert into the sorted register arrays (static indexing
    // only, so everything stays in VGPRs).
    auto insert = [&](float d2v, int node) {
        if (d2v < best[K_NN - 1]) {
            float cv = d2v;
            int   ci = node;
#pragma unroll
            for (int j2 = 0; j2 < K_NN; ++j2) {
                bool  sw = cv < best[j2];
                float nf = sw ? cv        : best[j2];
                float of = sw ? best[j2]  : cv;
                int   ni = sw ? ci        : bidx[j2];
                int   oi = sw ? bidx[j2]  : ci;
                best[j2] = nf; cv = of;
                bidx[j2] = ni; ci = oi;
            }
        }
    };

    float* mybuf = buf[wave];
    const int mb = hi ? 8 : 0;

    // Software-pipeline the B-operand fetch (nodes are WGP$/L2 resident).
    float4 nd = nodes[nn];

    for (int base = 0; base < N_NODES; base += 16) {
        // Prefetch next tile's node data; wrap keeps the address in-bounds on
        // the final iteration (value unused).
        float4 ndNext = nodes[((base + 16) & (N_NODES - 1)) + nn];

        // B operand (4x16 f32, K x N): lanes 0-15 rows K=0,1 (x,y);
        // lanes 16-31 rows K=2,3 (z, -0.5*||v||^2).
        v2f b;
        b[0] = hi ? nd.z : nd.x;
        b[1] = hi ? nd.w : nd.y;

        v8f c = {};
        v8f dLo = __builtin_amdgcn_wmma_f32_16x16x4_f32(
            false, aLo, false, b, (short)0, c, false, false);
        v8f dHi = __builtin_amdgcn_wmma_f32_16x16x4_f32(
            false, aHi, false, b, (short)0, c, false, false);

        // Stage D tiles to LDS: D vgpr r holds (M = r or r+8, N = lane&15).
#pragma unroll
        for (int r = 0; r < 8; ++r) mybuf[(mb + r) * 17 + nn]      = dLo[r];
#pragma unroll
        for (int r = 0; r < 8; ++r) mybuf[(16 + mb + r) * 17 + nn] = dHi[r];

        // Same-wave LDS ops execute in order on CDNA5 (DS pipe is in-order);
        // only compiler-level ordering must be pinned.
        asm volatile("" ::: "memory");

        const float* row = &mybuf[lane * 17];
        // Batch 4 candidate loads per iteration: 4 back-to-back ds_load_b32,
        // one DScnt wait, then 4 guarded inserts.
#pragma unroll 1
        for (int n4 = 0; n4 < 16; n4 += 4) {
            float c0 = row[n4 + 0];
            float c1 = row[n4 + 1];
            float c2 = row[n4 + 2];
            float c3 = row[n4 + 3];
            insert(fmaf(c0, -2.0f, p2), base + n4 + 0);
            insert(fmaf(c1, -2.0f, p2), base + n4 + 1);
            insert(fmaf(c2, -2.0f, p2), base + n4 + 2);
            insert(fmaf(c3, -2.0f, p2), base + n4 + 3);
        }
        asm volatile("" ::: "memory");

        nd = ndNext;
    }

    // ---- Weights ----
    const float invd = 1.0f / best[K_NN - 1];
    float wk[K_NN];
    float wsum = 0.0f;
#pragma unroll
    for (int k = 0; k < K_NN; ++k) {
        float r = 1.0f - best[k] * invd;
        r *= r;
        wk[k] = r;
        wsum += r;
    }
    const float invw = 1.0f / wsum;

    // ---- Blend (gathers are L2-resident: R/g/t total 240 KB) ----
    float pa0 = 0.f, pa1 = 0.f, pa2 = 0.f;
    float Ra[9];
#pragma unroll
    for (int i = 0; i < 9; ++i) Ra[i] = 0.f;

#pragma unroll
    for (int k = 0; k < K_NN; ++k) {
        const int j = bidx[k];
        const float w = wk[k] * invw;
        const float* Rj = Rm + j * 9;
        float r00 = Rj[0], r01 = Rj[1], r02 = Rj[2];
        float r10 = Rj[3], r11 = Rj[4], r12 = Rj[5];
        float r20 = Rj[6], r21 = Rj[7], r22 = Rj[8];
        float gx = g[3*j+0], gy = g[3*j+1], gz = g[3*j+2];
        float tx = t[3*j+0], ty = t[3*j+1], tz = t[3*j+2];

        float x0 = px - gx - tx;
        float x1 = py - gy - ty;
        float x2 = pz - gz - tz;

        // p_i = sum_j R[j][i] * x_j + g_i   (R^T x + g)
        float pb0 = r00*x0 + r10*x1 + r20*x2 + gx;
        float pb1 = r01*x0 + r11*x1 + r21*x2 + gy;
        float pb2 = r02*x0 + r12*x1 + r22*x2 + gz;
        pa0 += w * pb0;  pa1 += w * pb1;  pa2 += w * pb2;

        // R_blend[i][jj] += w * R[jj][i]  (weighted R^T)
        Ra[0] += w * r00;  Ra[1] += w * r10;  Ra[2] += w * r20;
        Ra[3] += w * r01;  Ra[4] += w * r11;  Ra[5] += w * r21;
        Ra[6] += w * r02;  Ra[7] += w * r12;  Ra[8] += w * r22;
    }

    if (best[0] > DIS_THRESH) pa0 = FAR_VAL;

    out_p[myPt*3 + 0] = pa0;
    out_p[myPt*3 + 1] = pa1;
    out_p[myPt*3 + 2] = pa2;
#pragma unroll
    for (int i = 0; i < 9; ++i) out_R[myPt*9 + i] = Ra[i];
}

extern "C" void kernel_launch(void* const* d_in, const int* in_sizes, int n_in,
                              void* d_out, int out_size, void* d_ws, size_t ws_size,
                              hipStream_t stream) {
    const float* inputs = (const float*)d_in[0];   // (65536, 35)
    const float* vd     = (const float*)d_in[1];   // (4096, 3)
    const float* Rm     = (const float*)d_in[2];   // (4096, 3, 3)
    const float* g      = (const float*)d_in[3];   // (4096, 3)
    const float* t      = (const float*)d_in[4];   // (4096, 3)
    float* out = (float*)d_out;                    // p_blend (3N) then R_blend (9N)

    float4* nodes = (float4*)d_ws;                 // 4096 * 16B = 64 KB scratch

    pack_nodes_kernel<<<N_NODES / 256, 256, 0, stream>>>(vd, nodes);
    deform_topk_blend_kernel<<<N_PTS / 256, 256, 0, stream>>>(
        inputs, nodes, Rm, g, t, out, out + (size_t)N_PTS * 3);
}